// Model_9534827397431
// MI455X (gfx1250) — compile-verified
//
#include <hip/hip_runtime.h>

// ============================================================================
// MI455X (gfx1250) implementation of the dual-GCN encoder + GRU decoder.
// All matmuls use V_WMMA_F32_16X16X4_F32 (exact f32, wave32, 16x16 tiles).
// Fast path: fully-interior tiles, K%8==0 -> unguarded loads, 2 WMMA / iter.
// Slow path: clamp+select (branchless) for ragged edges (O=3, K=3, P%16!=0).
// BatchNorm stats are deterministic two-stage reductions (no float atomics).
// Workspace requirement: ~289 MB (7*BIG + 2*MID + tail, see layout below).
// ============================================================================

typedef __attribute__((ext_vector_type(2))) float v2f;
typedef __attribute__((ext_vector_type(8))) float v8f;

#define LRELU_SLOPE 0.1f
#define BN_EPS 1e-5f

#define WMMA_F32(Acc, Af, Bf) \
  __builtin_amdgcn_wmma_f32_16x16x4_f32(false, (Af), false, (Bf), (short)0, (Acc), false, false)

// ---------------------------------------------------------------------------
// GEMM (channel-major): Out[n,o,p] (=|+=) sum_k Wk * X[n,k,p] (+ bias[o])
//   X: (N, K, P) contiguous per batch n.
//   W element: WT ? W[k*O + o] : W[o*K + k]
//   mode: 0 = store, 1 = accumulate into Out
// A-frag (32-bit 16x4): m = lane&15, k = vgpr + 2*(lane>=16)
// B-frag (32-bit 4x16): n = lane&15, k = vgpr + 2*(lane>=16)
// D: vgpr r -> m = r + 8*(lane>=16), col = lane&15
// ---------------------------------------------------------------------------
__global__ void gemm_cm_wmma(const float* __restrict__ X, const float* __restrict__ W,
                             const float* __restrict__ bias, float* __restrict__ Out,
                             int K, int P, int O, int WT, int mode) {
  const int lane = threadIdx.x & 31;
  const int half = lane >> 4;
  const int lm   = lane & 15;
  const int p0 = (blockIdx.x * blockDim.y + threadIdx.y) * 16;
  const int o0 = blockIdx.y * 16;
  const int n  = blockIdx.z;
  if (p0 >= P) return;                 // wave-uniform (EXEC stays all-1s)
  const float* x = X + (long)n * K * P;
  const int m   = o0 + lm;
  const int col = p0 + lm;
  v8f acc = {};
  const bool full = (o0 + 16 <= O) && (p0 + 16 <= P) && ((K & 7) == 0);
  if (full) {
    const float* xp = x + col;
    if (WT) {
      const float* wp = W + m;
      for (int k0 = 0; k0 < K; k0 += 8) {
        const int ka = k0 + 2 * half;
        v2f A0 = {wp[(long)ka * O],       wp[(long)(ka + 1) * O]};
        v2f B0 = {xp[(long)ka * P],       xp[(long)(ka + 1) * P]};
        v2f A1 = {wp[(long)(ka + 4) * O], wp[(long)(ka + 5) * O]};
        v2f B1 = {xp[(long)(ka + 4) * P], xp[(long)(ka + 5) * P]};
        acc = WMMA_F32(acc, A0, B0);
        acc = WMMA_F32(acc, A1, B1);
      }
    } else {
      const float* wp = W + (long)m * K;
      for (int k0 = 0; k0 < K; k0 += 8) {
        const int ka = k0 + 2 * half;          // even -> 8B aligned
        v2f A0 = *(const v2f*)(wp + ka);
        v2f B0 = {xp[(long)ka * P],       xp[(long)(ka + 1) * P]};
        v2f A1 = *(const v2f*)(wp + ka + 4);
        v2f B1 = {xp[(long)(ka + 4) * P], xp[(long)(ka + 5) * P]};
        acc = WMMA_F32(acc, A0, B0);
        acc = WMMA_F32(acc, A1, B1);
      }
    }
  } else {
    const int mc = (m < O) ? m : (O - 1);
    const int cc = (col < P) ? col : (P - 1);
    for (int k0 = 0; k0 < K; k0 += 4) {
      const int ka = k0 + 2 * half;
      const int kb = ka + 1;
      const int kac = (ka < K) ? ka : (K - 1);
      const int kbc = (kb < K) ? kb : (K - 1);
      float a0 = WT ? W[(long)kac * O + mc] : W[(long)mc * K + kac];
      float a1 = WT ? W[(long)kbc * O + mc] : W[(long)mc * K + kbc];
      float b0 = x[(long)kac * P + cc];
      float b1 = x[(long)kbc * P + cc];
      a0 = (m < O && ka < K) ? a0 : 0.f;
      a1 = (m < O && kb < K) ? a1 : 0.f;
      b0 = (col < P && ka < K) ? b0 : 0.f;
      b1 = (col < P && kb < K) ? b1 : 0.f;
      v2f A = {a0, a1};
      v2f B = {b0, b1};
      acc = WMMA_F32(acc, A, B);
    }
  }
  if (full) {
    float* op = Out + ((long)n * O + o0 + 8 * half) * P + col;
    for (int r = 0; r < 8; ++r) {
      float v = acc[r] + (bias ? bias[o0 + r + 8 * half] : 0.f);
      if (mode == 0) op[(long)r * P] = v; else op[(long)r * P] += v;
    }
  } else if (col < P) {
    for (int r = 0; r < 8; ++r) {
      const int mm = o0 + r + 8 * half;
      if (mm < O) {
        const long idx = ((long)n * O + mm) * P + col;
        float v = acc[r] + (bias ? bias[mm] : 0.f);
        if (mode == 0) Out[idx] = v; else Out[idx] += v;
      }
    }
  }
}

// ---------------------------------------------------------------------------
// Temporal-conv GEMM (implicit im2col): stride 2, kernel 7, pad 3.
//   Out[n,o,t',v] = sum_{c,kt} Wt[o,c,kt] * X[n,c,2t'+kt-3,v] + bias[o]
//   Wt: (O, Cin, 7) row-major, k = c*7+kt. Padding via clamp+select.
// ---------------------------------------------------------------------------
__global__ void gemm_tconv_wmma(const float* __restrict__ X, const float* __restrict__ Wt,
                                const float* __restrict__ bias, float* __restrict__ Out,
                                int Cin, int Tin, int O, int V) {
  const int Tout = Tin / 2;
  const int P = Tout * V;
  const int K = Cin * 7;
  const int lane = threadIdx.x & 31;
  const int half = lane >> 4;
  const int lm   = lane & 15;
  const int p0 = (blockIdx.x * blockDim.y + threadIdx.y) * 16;
  const int o0 = blockIdx.y * 16;
  const int n  = blockIdx.z;
  if (p0 >= P) return;
  const float* x = X + (long)n * Cin * Tin * V;
  const int m   = o0 + lm;
  const int col = p0 + lm;
  const int colc = (col < P) ? col : (P - 1);
  const int tp = colc / V;
  const int vv = colc - tp * V;
  const bool full = (o0 + 16 <= O) && (p0 + 16 <= P) && ((K & 7) == 0);

  auto bload = [&](int k) -> float {
    const int kc = (k < K) ? k : (K - 1);
    const int c  = kc / 7;
    const int kt = kc - c * 7;
    const int ti = 2 * tp + kt - 3;
    const int tic = (ti < 0) ? 0 : ((ti < Tin) ? ti : (Tin - 1));
    float v = x[((long)c * Tin + tic) * V + vv];
    return (ti >= 0 && ti < Tin && k < K && col < P) ? v : 0.f;
  };

  v8f acc = {};
  if (full) {
    const float* wp = Wt + (long)m * K;
    for (int k0 = 0; k0 < K; k0 += 8) {
      const int ka = k0 + 2 * half;
      v2f A0 = *(const v2f*)(wp + ka);
      v2f B0 = {bload(ka), bload(ka + 1)};
      v2f A1 = *(const v2f*)(wp + ka + 4);
      v2f B1 = {bload(ka + 4), bload(ka + 5)};
      acc = WMMA_F32(acc, A0, B0);
      acc = WMMA_F32(acc, A1, B1);
    }
  } else {
    const int mc = (m < O) ? m : (O - 1);
    for (int k0 = 0; k0 < K; k0 += 4) {
      const int ka = k0 + 2 * half;
      const int kb = ka + 1;
      const int kac = (ka < K) ? ka : (K - 1);
      const int kbc = (kb < K) ? kb : (K - 1);
      float a0 = Wt[(long)mc * K + kac];
      float a1 = Wt[(long)mc * K + kbc];
      a0 = (m < O && ka < K) ? a0 : 0.f;
      a1 = (m < O && kb < K) ? a1 : 0.f;
      v2f A = {a0, a1};
      v2f B = {bload(ka), bload(kb)};
      acc = WMMA_F32(acc, A, B);
    }
  }
  if (full) {
    float* op = Out + ((long)n * O + o0 + 8 * half) * P + col;
    for (int r = 0; r < 8; ++r)
      op[(long)r * P] = acc[r] + (bias ? bias[o0 + r + 8 * half] : 0.f);
  } else if (col < P) {
    for (int r = 0; r < 8; ++r) {
      const int mm = o0 + r + 8 * half;
      if (mm < O)
        Out[((long)n * O + mm) * P + col] = acc[r] + (bias ? bias[mm] : 0.f);
    }
  }
}

// ---------------------------------------------------------------------------
// GEMM (row-major): Out[r,o] (=|+=) sum_k X[r,k] * W[k,o] (+ bias[o])
//   mode: 0 = store, 1 = accumulate, 2 = store with leaky-ReLU(0.1)
// ---------------------------------------------------------------------------
__global__ void gemm_rm_wmma(const float* __restrict__ X, const float* __restrict__ W,
                             const float* __restrict__ bias, float* __restrict__ Out,
                             int R, int K, int O, int mode) {
  const int lane = threadIdx.x & 31;
  const int half = lane >> 4;
  const int lm   = lane & 15;
  const int o0 = blockIdx.x * 16;
  const int r0 = (blockIdx.y * blockDim.y + threadIdx.y) * 16;
  if (r0 >= R) return;
  const int row = r0 + lm;
  const int col = o0 + lm;
  v8f acc = {};
  const bool full = (r0 + 16 <= R) && (o0 + 16 <= O) && ((K & 7) == 0);
  if (full) {
    const float* xp = X + (long)row * K;
    const float* wp = W + col;
    for (int k0 = 0; k0 < K; k0 += 8) {
      const int ka = k0 + 2 * half;
      v2f A0 = *(const v2f*)(xp + ka);
      v2f B0 = {wp[(long)ka * O],       wp[(long)(ka + 1) * O]};
      v2f A1 = *(const v2f*)(xp + ka + 4);
      v2f B1 = {wp[(long)(ka + 4) * O], wp[(long)(ka + 5) * O]};
      acc = WMMA_F32(acc, A0, B0);
      acc = WMMA_F32(acc, A1, B1);
    }
  } else {
    const int rowc = (row < R) ? row : (R - 1);
    const int colc = (col < O) ? col : (O - 1);
    for (int k0 = 0; k0 < K; k0 += 4) {
      const int ka = k0 + 2 * half;
      const int kb = ka + 1;
      const int kac = (ka < K) ? ka : (K - 1);
      const int kbc = (kb < K) ? kb : (K - 1);
      float a0 = X[(long)rowc * K + kac];
      float a1 = X[(long)rowc * K + kbc];
      float b0 = W[(long)kac * O + colc];
      float b1 = W[(long)kbc * O + colc];
      a0 = (row < R && ka < K) ? a0 : 0.f;
      a1 = (row < R && kb < K) ? a1 : 0.f;
      b0 = (col < O && ka < K) ? b0 : 0.f;
      b1 = (col < O && kb < K) ? b1 : 0.f;
      v2f A = {a0, a1};
      v2f B = {b0, b1};
      acc = WMMA_F32(acc, A, B);
    }
  }
  if (col < O) {
    const float bb = bias ? bias[col] : 0.f;
    for (int r = 0; r < 8; ++r) {
      const int rr = r0 + r + 8 * half;
      if (rr < R) {
        float v = acc[r] + bb;
        if (mode == 2) v = (v >= 0.f) ? v : LRELU_SLOPE * v;
        const long idx = (long)rr * O + col;
        if (mode == 1) Out[idx] += v; else Out[idx] = v;
      }
    }
  }
}

// ---------------------------------------------------------------------------
// BatchNorm stage 1: per (channel, chunk) partial sum / sumsq. Deterministic.
// ---------------------------------------------------------------------------
#define BN_CHUNKS 32
__global__ void bn_stats_partial(const float* __restrict__ X, float* __restrict__ partial,
                                 int C, int Nb, int P) {
  const int c = blockIdx.x;
  const long start  = (long)blockIdx.y * blockDim.x + threadIdx.x;
  const long stride = (long)gridDim.y * blockDim.x;
  float s = 0.f, ss = 0.f;
  for (int nn = 0; nn < Nb; ++nn) {
    const float* xp = X + ((long)nn * C + c) * P;
    for (long p = start; p < P; p += stride) {
      const float v = xp[p];
      s += v; ss += v * v;
    }
  }
  __shared__ float sh1[256], sh2[256];
  const int t = threadIdx.x;
  sh1[t] = s; sh2[t] = ss;
  __syncthreads();
  for (int o = 128; o > 0; o >>= 1) {
    if (t < o) { sh1[t] += sh1[t + o]; sh2[t] += sh2[t + o]; }
    __syncthreads();
  }
  if (t == 0) {
    partial[((long)c * gridDim.y + blockIdx.y) * 2 + 0] = sh1[0];
    partial[((long)c * gridDim.y + blockIdx.y) * 2 + 1] = sh2[0];
  }
}

__global__ void bn_stats_reduce(const float* __restrict__ partial, float* __restrict__ stats,
                                int C) {
  const int c = blockIdx.x * blockDim.x + threadIdx.x;
  if (c >= C) return;
  float s = 0.f, ss = 0.f;
  for (int i = 0; i < BN_CHUNKS; ++i) {
    s  += partial[((long)c * BN_CHUNKS + i) * 2 + 0];
    ss += partial[((long)c * BN_CHUNKS + i) * 2 + 1];
  }
  stats[2 * c]     = s;
  stats[2 * c + 1] = ss;
}

// BatchNorm apply: dst (=|+=) (src - mean) * rsqrt(var+eps) * g + be [optional ReLU]
__global__ void bn_apply(const float* __restrict__ src, float* __restrict__ dst,
                         const float* __restrict__ stats, const float* __restrict__ g,
                         const float* __restrict__ be, int C, int P, long total,
                         float invCnt, int relu, int accum) {
  const long idx = (long)blockIdx.x * blockDim.x + threadIdx.x;
  if (idx >= total) return;
  const int c = (int)((idx / P) % C);
  const float m   = stats[2 * c] * invCnt;
  const float var = stats[2 * c + 1] * invCnt - m * m;
  float y = (src[idx] - m) * rsqrtf(var + BN_EPS) * g[c] + be[c];
  if (relu) y = fmaxf(y, 0.f);
  if (accum) dst[idx] += y; else dst[idx] = y;
}

// V-adjacency mix: out[row,w] = sum_v in[row,v] * (trans ? A[w,v] : A[v,w]); V=25
__global__ void vmix(const float* __restrict__ in, const float* __restrict__ A,
                     float* __restrict__ out, long rows, int V, int trans) {
  const long idx = (long)blockIdx.x * blockDim.x + threadIdx.x;
  if (idx >= rows * V) return;
  const long row = idx / V;
  const int  w   = (int)(idx - row * V);
  const float* xi = in + row * V;
  float s = 0.f;
  for (int v = 0; v < V; ++v)
    s += xi[v] * (trans ? A[(long)w * V + v] : A[(long)v * V + w]);
  out[idx] = s;
}

// Decoder init: ld[n,u,d] = ls[n,d,u] - lss[n,d,u]; lst[n,u,d] = ls[n,d,u]; U=50
__global__ void dec_init(const float* __restrict__ ls, const float* __restrict__ lss,
                         float* __restrict__ ld, float* __restrict__ lst, int N, int U) {
  const long idx = (long)blockIdx.x * blockDim.x + threadIdx.x;
  if (idx >= (long)N * U * 3) return;
  const int n = (int)(idx / (U * 3));
  const int rem = (int)(idx - (long)n * U * 3);
  const int u = rem / 3, d = rem - u * 3;
  const long src = ((long)n * 3 + d) * U + u;
  ld[idx]  = ls[src] - lss[src];
  lst[idx] = ls[src];
}

// H[n,u,c] = (u<25 ? jH2[n,c,0,u] : bH2[n,c,0,u-25]); H: (N,50,256)
__global__ void build_H(const float* __restrict__ jH2, const float* __restrict__ bH2,
                        float* __restrict__ H, int N) {
  const long idx = (long)blockIdx.x * blockDim.x + threadIdx.x;
  if (idx >= (long)N * 50 * 256) return;
  const int n = (int)(idx / (50 * 256));
  const int rem = (int)(idx - (long)n * 50 * 256);
  const int u = rem / 256, c = rem - u * 256;
  if (u < 25) H[idx] = jH2[((long)n * 256 + c) * 25 + u];
  else        H[idx] = bH2[((long)n * 256 + c) * 25 + (u - 25)];
}

// jH[n,c,v] = H[n,v,c]; bH[n,c,v] = H[n,25+v,c]
__global__ void split_H(const float* __restrict__ H, float* __restrict__ jH,
                        float* __restrict__ bH, int N) {
  const long idx = (long)blockIdx.x * blockDim.x + threadIdx.x;
  if (idx >= (long)N * 256 * 25) return;
  const int n = (int)(idx / (256 * 25));
  const int rem = (int)(idx - (long)n * 256 * 25);
  const int c = rem / 25, v = rem - c * 25;
  jH[idx] = H[((long)n * 50 + v) * 256 + c];
  bH[idx] = H[((long)n * 50 + 25 + v) * 256 + c];
}

// GRU combine: r=sig(Xr); z=sig(Xz); n=tanh(Xn + r*Hh); H = (1-z)*n + z*H
__global__ void gru_combine(const float* __restrict__ Xr, const float* __restrict__ Xz,
                            const float* __restrict__ Xn, const float* __restrict__ Hh,
                            float* __restrict__ H, long total) {
  const long idx = (long)blockIdx.x * blockDim.x + threadIdx.x;
  if (idx >= total) return;
  const float r = 1.f / (1.f + expf(-Xr[idx]));
  const float z = 1.f / (1.f + expf(-Xz[idx]));
  const float n = tanhf(Xn[idx] + r * Hh[idx]);
  H[idx] = (1.f - z) * n + z * H[idx];
}

// ld += pred; lst += ld; emit out[n, u(<25), step, d] = lst
__global__ void emit_step(const float* __restrict__ pred, float* __restrict__ ld,
                          float* __restrict__ lst, float* __restrict__ out,
                          int N, int step) {
  const long idx = (long)blockIdx.x * blockDim.x + threadIdx.x;
  if (idx >= (long)N * 50 * 3) return;
  const int n = (int)(idx / 150);
  const int rem = (int)(idx - (long)n * 150);
  const int u = rem / 3, d = rem - u * 3;
  const float ldv = ld[idx] + pred[idx];
  ld[idx] = ldv;
  const float s = lst[idx] + ldv;
  lst[idx] = s;
  if (u < 25) out[(((long)n * 25 + u) * 10 + step) * 3 + d] = s;
}

// ============================================================================
// Host orchestration
// ============================================================================
extern "C" void kernel_launch(void* const* d_in, const int* in_sizes, int n_in,
                              void* d_out, int out_size, void* d_ws, size_t ws_size,
                              hipStream_t stream) {
  const int N = 128, V = 25;
  const float* j_inputs   = (const float*)d_in[0];
  const float* b_inputs   = (const float*)d_in[1];
  const float* last_state = (const float*)d_in[2];
  const float* last_sec   = (const float*)d_in[3];

  // ---- parameter cursor (depth-first dict order) --------------------------
  int ci = 4;
  auto nx = [&]() { return (const float*)d_in[ci++]; };
  struct GcnP { const float *A, *w, *b, *g, *be; };
  struct UpP  { const float *As, *At, *Wv, *bv, *gv, *bev; };
  struct TcP  { const float *w, *b, *g, *be; };
  struct BlkP { GcnP gj, gb; UpP up; TcP tc, res; bool has_res; };
  auto rdG = [&]() { GcnP g; g.A = nx(); g.w = nx(); g.b = nx(); g.g = nx(); g.be = nx(); return g; };
  auto rdU = [&]() { UpP u; u.As = nx(); u.At = nx(); u.Wv = nx(); u.bv = nx(); u.gv = nx(); u.bev = nx(); return u; };
  auto rdT = [&]() { TcP t; t.w = nx(); t.b = nx(); t.g = nx(); t.be = nx(); return t; };
  BlkP blk[6];
  for (int i = 0; i < 6; ++i) {
    blk[i].gj = rdG(); blk[i].gb = rdG(); blk[i].up = rdU(); blk[i].tc = rdT();
    blk[i].has_res = (i > 0);
    if (i > 0) blk[i].res = rdT();
  }
  GcnP fin_gj = rdG(), fin_gb = rdG();
  UpP  fin_up = rdU();
  const float *Wir = nx(), *bir = nx(), *Wii = nx(), *bii = nx(), *Win = nx(), *bin = nx();
  const float *Whr = nx(), *Whi = nx(), *Whh = nx();
  const float *W1 = nx(), *c1 = nx(), *W2 = nx(), *c2 = nx(), *W3 = nx(), *c3 = nx();

  // ---- workspace layout (floats) ------------------------------------------
  const long BIG = 128L * 2048 * 25;   //  6,553,600 (max C*T activation)
  const long MID = 128L * 4096 * 25;   // 13,107,200 (max Cout*Tin update buf)
  float* ws = (float*)d_ws;
  float* A_j = ws;                     // ping-pong pair A
  float* A_b = ws + BIG;
  float* fj  = ws + 2 * BIG;
  float* fb  = ws + 3 * BIG;
  float* tmp = ws + 4 * BIG;
  float* ju  = ws + 5 * BIG;           // size MID
  float* bu  = ws + 5 * BIG + MID;     // size MID
  float* B_j = ws + 5 * BIG + 2 * MID; // ping-pong pair B
  float* B_b = B_j + BIG;
  float* stats   = B_b + BIG;          // 512
  float* ld      = stats + 512;        // N*50*3
  float* lst     = ld + (long)N * 150;
  float* pred    = lst + (long)N * 150;
  float* partial = pred + (long)N * 150;  // 2*C*BN_CHUNKS <= 16384

  // decoder carve-outs (regions dead at that point)
  const long SM = (long)N * 256 * 25;  // 819,200
  const long DH = (long)N * 50 * 256;  // 1,638,400
  float* H  = ju;          // ju region (7*DH < MID)
  float* Xr = ju + DH;
  float* Xz = ju + 2 * DH;
  float* Xn = ju + 3 * DH;
  float* Hh = ju + 4 * DH;
  float* h1 = ju + 5 * DH;
  float* h2 = ju + 6 * DH;
  float* fjD   = B_j;      // B pair dead after encoder (final outputs in A)
  float* fbD   = B_j + SM;
  float* gtmpD = B_j + 2 * SM;
  float* mtD   = B_j + 3 * SM;
  float* jH2D  = B_j + 4 * SM;
  float* bH2D  = B_j + 5 * SM;
  float* jHs   = B_b;
  float* bHs   = B_b + SM;

  // ---- launch helpers ------------------------------------------------------
  auto gemmCM = [&](const float* X, const float* W, const float* bias, float* Out,
                    int K, int P, int O, int WT, int mode) {
    dim3 bl(32, 4), gr((P + 63) / 64, (O + 15) / 16, N);
    gemm_cm_wmma<<<gr, bl, 0, stream>>>(X, W, bias, Out, K, P, O, WT, mode);
  };
  auto tconvL = [&](const float* X, const float* Wt, const float* bias, float* Out,
                    int Cin, int Tin, int O) {
    const int P = (Tin / 2) * V;
    dim3 bl(32, 4), gr((P + 63) / 64, (O + 15) / 16, N);
    gemm_tconv_wmma<<<gr, bl, 0, stream>>>(X, Wt, bias, Out, Cin, Tin, O, V);
  };
  auto gemmRM = [&](const float* X, const float* W, const float* bias, float* Out,
                    int R, int K, int O, int mode) {
    dim3 bl(32, 4), gr((O + 15) / 16, (R + 63) / 64, 1);
    gemm_rm_wmma<<<gr, bl, 0, stream>>>(X, W, bias, Out, R, K, O, mode);
  };
  auto bnrun = [&](const float* src, float* dst, int C, int P, const float* g,
                   const float* be, int relu, int accum) {
    bn_stats_partial<<<dim3(C, BN_CHUNKS), 256, 0, stream>>>(src, partial, C, N, P);
    bn_stats_reduce<<<(C + 63) / 64, 64, 0, stream>>>(partial, stats, C);
    const long total = (long)N * C * P;
    bn_apply<<<(unsigned)((total + 255) / 256), 256, 0, stream>>>(
        src, dst, stats, g, be, C, P, total, 1.0f / (float)((long)N * P), relu, accum);
  };
  auto vmixL = [&](const float* in, const float* A, float* out, long rows, int trans) {
    const long total = rows * V;
    vmix<<<(unsigned)((total + 255) / 256), 256, 0, stream>>>(in, A, out, rows, V, trans);
  };

  // ---- one encoder block ---------------------------------------------------
  auto run_block = [&](const float* inj, const float* inb, float* outj, float* outb,
                       const BlkP& p, int Cin, int Cout, int Tin) {
    const int  P    = Tin * V;
    const int  Pout = (Tin / 2) * V;
    const long rows = (long)N * Cin * Tin;
    const long wvO  = (long)Cin * Cout;
    // GCN(j): W-mix -> BN -> A-mix
    gemmCM(inj, p.gj.w, p.gj.b, tmp, Cin, P, Cin, 0, 0);
    bnrun(tmp, tmp, Cin, P, p.gj.g, p.gj.be, 0, 0);
    vmixL(tmp, p.gj.A, fj, rows, 0);
    // GCN(b)
    gemmCM(inb, p.gb.w, p.gb.b, tmp, Cin, P, Cin, 0, 0);
    bnrun(tmp, tmp, Cin, P, p.gb.g, p.gb.be, 0, 0);
    vmixL(tmp, p.gb.A, fb, rows, 0);
    // update(j): [fj, fb@As, fb@At] @ Wv  (3 accumulated GEMMs)
    gemmCM(fj, p.up.Wv, p.up.bv, ju, Cin, P, Cout, 1, 0);
    vmixL(fb, p.up.As, tmp, rows, 0);
    gemmCM(tmp, p.up.Wv + wvO, nullptr, ju, Cin, P, Cout, 1, 1);
    vmixL(fb, p.up.At, tmp, rows, 0);
    gemmCM(tmp, p.up.Wv + 2 * wvO, nullptr, ju, Cin, P, Cout, 1, 1);
    bnrun(ju, ju, Cout, P, p.up.gv, p.up.bev, 1, 0);
    // update(b): [fb, fj@As.T, fj@At.T] @ Wv
    gemmCM(fb, p.up.Wv, p.up.bv, bu, Cin, P, Cout, 1, 0);
    vmixL(fj, p.up.As, tmp, rows, 1);
    gemmCM(tmp, p.up.Wv + wvO, nullptr, bu, Cin, P, Cout, 1, 1);
    vmixL(fj, p.up.At, tmp, rows, 1);
    gemmCM(tmp, p.up.Wv + 2 * wvO, nullptr, bu, Cin, P, Cout, 1, 1);
    bnrun(bu, bu, Cout, P, p.up.gv, p.up.bev, 1, 0);
    // temporal conv + BN (+ residual conv + BN, accumulated)
    tconvL(ju, p.tc.w, p.tc.b, outj, Cout, Tin, Cout);
    bnrun(outj, outj, Cout, Pout, p.tc.g, p.tc.be, 0, 0);
    tconvL(bu, p.tc.w, p.tc.b, outb, Cout, Tin, Cout);
    bnrun(outb, outb, Cout, Pout, p.tc.g, p.tc.be, 0, 0);
    if (p.has_res) {
      tconvL(inj, p.res.w, p.res.b, tmp, Cin, Tin, Cout);
      bnrun(tmp, outj, Cout, Pout, p.res.g, p.res.be, 0, 1);
      tconvL(inb, p.res.w, p.res.b, tmp, Cin, Tin, Cout);
      bnrun(tmp, outb, Cout, Pout, p.res.g, p.res.be, 0, 1);
    }
  };

  // ---- encoder: b1..b6, b6 again -------------------------------------------
  const int Cs[8] = {3, 32, 64, 128, 128, 256, 256, 256};
  const int Ts[8] = {128, 64, 32, 16, 8, 4, 2, 1};
  const float* cj = j_inputs;
  const float* cb = b_inputs;
  for (int i = 0; i < 7; ++i) {
    const BlkP& p = blk[(i < 6) ? i : 5];
    float* oj = (i % 2 == 0) ? A_j : B_j;
    float* ob = (i % 2 == 0) ? A_b : B_b;
    run_block(cj, cb, oj, ob, p, Cs[i], Cs[i + 1], Ts[i]);
    cj = oj; cb = ob;
  }
  // final outputs (N,256,1,25) now in A pair (cj/cb)

  // ---- decoder -------------------------------------------------------------
  const long RU = (long)N * 50;
  dec_init<<<(unsigned)((RU * 3 + 255) / 256), 256, 0, stream>>>(last_state, last_sec,
                                                                 ld, lst, N, 50);
  const float* jH = cj;
  const float* bH = cb;
  const long wo = 256L * 256;
  for (int step = 0; step < 10; ++step) {
    if (step > 0) {
      split_H<<<(unsigned)((SM + 255) / 256), 256, 0, stream>>>(H, jHs, bHs, N);
      jH = jHs; bH = bHs;
    }
    // fin _da_gco (T=1, P=25, C=256)
    gemmCM(jH, fin_gj.w, fin_gj.b, gtmpD, 256, 25, 256, 0, 0);
    bnrun(gtmpD, gtmpD, 256, 25, fin_gj.g, fin_gj.be, 0, 0);
    vmixL(gtmpD, fin_gj.A, fjD, (long)N * 256, 0);
    gemmCM(bH, fin_gb.w, fin_gb.b, gtmpD, 256, 25, 256, 0, 0);
    bnrun(gtmpD, gtmpD, 256, 25, fin_gb.g, fin_gb.be, 0, 0);
    vmixL(gtmpD, fin_gb.A, fbD, (long)N * 256, 0);
    gemmCM(fjD, fin_up.Wv, fin_up.bv, jH2D, 256, 25, 256, 1, 0);
    vmixL(fbD, fin_up.As, mtD, (long)N * 256, 0);
    gemmCM(mtD, fin_up.Wv + wo, nullptr, jH2D, 256, 25, 256, 1, 1);
    vmixL(fbD, fin_up.At, mtD, (long)N * 256, 0);
    gemmCM(mtD, fin_up.Wv + 2 * wo, nullptr, jH2D, 256, 25, 256, 1, 1);
    bnrun(jH2D, jH2D, 256, 25, fin_up.gv, fin_up.bev, 1, 0);
    gemmCM(fbD, fin_up.Wv, fin_up.bv, bH2D, 256, 25, 256, 1, 0);
    vmixL(fjD, fin_up.As, mtD, (long)N * 256, 1);
    gemmCM(mtD, fin_up.Wv + wo, nullptr, bH2D, 256, 25, 256, 1, 1);
    vmixL(fjD, fin_up.At, mtD, (long)N * 256, 1);
    gemmCM(mtD, fin_up.Wv + 2 * wo, nullptr, bH2D, 256, 25, 256, 1, 1);
    bnrun(bH2D, bH2D, 256, 25, fin_up.gv, fin_up.bev, 1, 0);
    build_H<<<(unsigned)((DH + 255) / 256), 256, 0, stream>>>(jH2D, bH2D, H, N);
    // GRU
    gemmRM(ld, Wir, bir, Xr, (int)RU, 3, 256, 0);
    gemmRM(H, Whr, nullptr, Xr, (int)RU, 256, 256, 1);
    gemmRM(ld, Wii, bii, Xz, (int)RU, 3, 256, 0);
    gemmRM(H, Whi, nullptr, Xz, (int)RU, 256, 256, 1);
    gemmRM(ld, Win, bin, Xn, (int)RU, 3, 256, 0);
    gemmRM(H, Whh, nullptr, Hh, (int)RU, 256, 256, 0);
    gru_combine<<<(unsigned)((DH + 255) / 256), 256, 0, stream>>>(Xr, Xz, Xn, Hh, H, DH);
    // MLP head (leaky ReLU fused in epilogue)
    gemmRM(H, W1, c1, h1, (int)RU, 256, 256, 2);
    gemmRM(h1, W2, c2, h2, (int)RU, 256, 256, 2);
    gemmRM(h2, W3, c3, pred, (int)RU, 256, 3, 0);
    emit_step<<<(unsigned)((RU * 3 + 255) / 256), 256, 0, stream>>>(pred, ld, lst,
                                                                    (float*)d_out, N, step);
  }
  (void)in_sizes; (void)n_in; (void)out_size; (void)ws_size;
}